// BlockSparseLinear_30356828848770
// MI455X (gfx1250) — compile-verified
//
#include <hip/hip_runtime.h>
#include <hip/hip_bf16.h>

#define BS      32
#define RB      128
#define MAXNNZ  13
#define BATCH   2048
#define XCOLS   4096      // CB*BS
#define YCOLS   4096      // RB*BS
#define WPAD    33        // padded LDS row stride (floats) -> no bank conflicts
#define WBLK    (BS * WPAD)

typedef __attribute__((ext_vector_type(2))) float v2f;
typedef __attribute__((ext_vector_type(8))) float v8f;

__global__ __launch_bounds__(256)
void BlockSparseLinear_kernel(const float* __restrict__ x,
                              const int*   __restrict__ crow,
                              const int*   __restrict__ cols,
                              const float* __restrict__ mask,
                              const float* __restrict__ weight,
                              const float* __restrict__ bias,
                              float*       __restrict__ y)
{
    __shared__ float wlds[MAXNNZ * WBLK];   // 13*1056*4 = 54,912 bytes

    const int r         = blockIdx.y;
    const int row_start = crow[r];
    int nnz = crow[r + 1] - row_start;
    if (nnz > MAXNNZ) nnz = MAXNNZ;

    // ---- stage masked weights into LDS (w[o][s] at padded stride) ----
    const int tid = threadIdx.x;
    const long wbase = (long)row_start * (BS * BS);
    for (int i = tid; i < nnz * BS * BS; i += 256) {
        const int j   = i >> 10;        // block within row
        const int rem = i & 1023;
        const int o   = rem >> 5;
        const int s   = rem & 31;
        wlds[j * WBLK + o * WPAD + s] = weight[wbase + i] * mask[wbase + i];
    }
    __syncthreads();

    // ---- per-wave tile assignment ----
    const int wave = tid >> 5;          // 0..7
    const int lane = tid & 31;
    const int lm   = lane & 15;         // M / N index within tile
    const int kh   = lane >> 4;         // K-half select (0 -> K0/K1, 1 -> K2/K3)
    const int mw   = wave >> 1;         // 0..3 : 32-row batch sub-tile
    const int nw   = wave & 1;          // 0..1 : 16-col output sub-tile

    const int m0 = blockIdx.x * 128 + mw * 32;
    const int no = nw * 16;

    v8f c0 = {};
    v8f c1 = {};

    const float* __restrict__ xrow0 = x + (long)(m0 + lm) * XCOLS;
    const float* __restrict__ xrow1 = x + (long)(m0 + 16 + lm) * XCOLS;

    for (int j = 0; j < nnz; ++j) {
        const int cb = cols[row_start + j];
        const float* __restrict__ xa0 = xrow0 + cb * BS;
        const float* __restrict__ xa1 = xrow1 + cb * BS;
        const float* __restrict__ wb  = &wlds[j * WBLK + (no + lm) * WPAD];
#pragma unroll
        for (int k = 0; k < 8; ++k) {
            const int base = k * 4 + kh * 2;
            // B fragment: B[K][N] = w[o=N][s=K]; lanes hold N, kh selects K pair
            v2f b;
            b.x = wb[base + 0];
            b.y = wb[base + 1];
            // A fragments: lanes hold M, kh selects K pair (16x4 f32 layout)
            v2f a0, a1;
            a0.x = xa0[base + 0];
            a0.y = xa0[base + 1];
            a1.x = xa1[base + 0];
            a1.y = xa1[base + 1];
            c0 = __builtin_amdgcn_wmma_f32_16x16x4_f32(false, a0, false, b,
                                                       (short)0, c0, false, false);
            c1 = __builtin_amdgcn_wmma_f32_16x16x4_f32(false, a1, false, b,
                                                       (short)0, c1, false, false);
        }
    }

    // ---- epilogue: bias + store (D layout: VGPR g -> rows g and g+8) ----
    const int   colg = r * BS + no + lm;
    const float bv   = bias[colg];
#pragma unroll
    for (int g = 0; g < 8; ++g) {
        const int row0 = m0 + g + kh * 8;
        y[(long)row0        * YCOLS + colg] = c0[g] + bv;
        y[(long)(row0 + 16) * YCOLS + colg] = c1[g] + bv;
    }
}

extern "C" void kernel_launch(void* const* d_in, const int* in_sizes, int n_in,
                              void* d_out, int out_size, void* d_ws, size_t ws_size,
                              hipStream_t stream)
{
    const float* x      = (const float*)d_in[0];
    const int*   crow   = (const int*)  d_in[1];
    const int*   cols   = (const int*)  d_in[2];
    const float* mask   = (const float*)d_in[3];
    const float* weight = (const float*)d_in[4];
    const float* bias   = (const float*)d_in[5];
    float*       y      = (float*)d_out;

    dim3 grid(BATCH / 128, RB, 1);   // 16 x 128 workgroups
    dim3 block(256, 1, 1);           // 8 waves (wave32)
    BlockSparseLinear_kernel<<<grid, block, 0, stream>>>(x, crow, cols, mask,
                                                         weight, bias, y);
}